// TrafficGAT_45354854645879
// MI455X (gfx1250) — compile-verified
//
#include <hip/hip_runtime.h>
#include <math.h>

// TrafficGAT on MI455X (gfx1250, wave32).
// Memory-bound graph workload: dense node GEMMs use V_WMMA_F32_16X16X4_F32
// (exact fp32, matches reference numerics) with A tiles staged via
// GLOBAL_LOAD_ASYNC_TO_LDS_B128 (ASYNCcnt); gather/scatter phases use fp32
// global atomics. Workspace requirement ~322 MB (layout below).
// NOTE: float atomicAdd scatter order is hardware-scheduled; results are
// correct to fp32 reduction-order tolerance (same caveat as any segment_sum).

#define HID 64
#define HEADS 4
#define NEG_SLOPE 0.2f

typedef __attribute__((ext_vector_type(2))) float v2f;
typedef __attribute__((ext_vector_type(8))) float v8f;

__device__ __forceinline__ float leaky(float x) { return x > 0.0f ? x : NEG_SLOPE * x; }

// float atomic-max via signed-max / unsigned-min bit trick (init must be -inf)
__device__ __forceinline__ void atomicMaxF(float* addr, float val) {
  if (val >= 0.0f) atomicMax((int*)addr, __float_as_int(val));
  else             atomicMin((unsigned int*)addr, __float_as_uint(val));
}

// ---------------- utility: fill ----------------
__global__ void k_fill(float* p, float v, long long n) {
  long long i = (long long)blockIdx.x * blockDim.x + threadIdx.x;
  long long stride = (long long)gridDim.x * blockDim.x;
  for (; i < n; i += stride) p[i] = v;
}

// ---------------- edge-attn scalar constants ----------------
// EDGE_DIM==1 => al_e[e,h] = edge_attr[e] * dot(We[head h], ae[h]) = ea[e]*ce[h]
__global__ void k_ce(const float* We1, const float* ae1, const float* We2,
                     const float* ae2, float* ce) {
  int t = threadIdx.x;
  if (t < HEADS) {
    float s = 0.f;
    for (int i = 0; i < HID; ++i) s += We1[t * HID + i] * ae1[t * HID + i];
    ce[t] = s;
  } else if (t == HEADS) {
    float s = 0.f;
    for (int i = 0; i < HID; ++i) s += We2[i] * ae2[i];
    ce[HEADS] = s;
  }
}

// ---------------- bias row: bb[n] = sum_k bias[k] * B[k,n]  (n < 64) --------
// Lets the GEMM A-stage be a pure byte copy: (A + 1*bias^T) @ B = A@B + 1*(bias^T@B).
__global__ void k_biasrow(const float* __restrict__ bias, const float* __restrict__ B,
                          float* __restrict__ bb, int K) {
  int n = threadIdx.x;  // 64
  float s = 0.f;
  for (int k = 0; k < K; ++k) s += bias[k] * B[(size_t)k * 64 + n];
  bb[n] = s;
}

// ---------------- layer-1 node transform ----------------
// h1[n,256] = x[n,2] @ W1[2,256];  als1[n,h]=dot(h1 head h, as1[h]); ald1 likewise.
__global__ void k_node1(const float* __restrict__ x, const float* __restrict__ W1,
                        const float* __restrict__ as1, const float* __restrict__ ad1,
                        float* __restrict__ h1, float* __restrict__ als1,
                        float* __restrict__ ald1) {
  int n = blockIdx.x;
  int c = threadIdx.x;  // 256 = HEADS*HID channels
  float x0 = x[n * 2 + 0], x1 = x[n * 2 + 1];
  float v = x0 * W1[c] + x1 * W1[256 + c];
  h1[(size_t)n * 256 + c] = v;
  __shared__ float ss[256];
  __shared__ float sd[256];
  ss[c] = v * as1[c];
  sd[c] = v * ad1[c];
  __syncthreads();
  if (c < HEADS) {
    float a = 0.f, b = 0.f;
    for (int i = 0; i < HID; ++i) { a += ss[c * HID + i]; b += sd[c * HID + i]; }
    als1[n * HEADS + c] = a;
    ald1[n * HEADS + c] = b;
  }
}

// ---------------- layer-1 edge logits + segment max over dst ----------------
__global__ void k_logits1(const int* __restrict__ ei, const float* __restrict__ ea,
                          const float* __restrict__ als1, const float* __restrict__ ald1,
                          const float* __restrict__ ce, float* __restrict__ lg1,
                          float* __restrict__ m1, int E) {
  int e = blockIdx.x * blockDim.x + threadIdx.x;
  if (e >= E) return;
  int s = ei[e], d = ei[E + e];
  float a = ea[e];
  float4 vs = *(const float4*)(als1 + (size_t)s * 4);
  float4 vd = *(const float4*)(ald1 + (size_t)d * 4);
  float4 lg;
  lg.x = leaky(vs.x + vd.x + a * ce[0]);
  lg.y = leaky(vs.y + vd.y + a * ce[1]);
  lg.z = leaky(vs.z + vd.z + a * ce[2]);
  lg.w = leaky(vs.w + vd.w + a * ce[3]);
  *(float4*)(lg1 + (size_t)e * 4) = lg;
  atomicMaxF(&m1[d * 4 + 0], lg.x);
  atomicMaxF(&m1[d * 4 + 1], lg.y);
  atomicMaxF(&m1[d * 4 + 2], lg.z);
  atomicMaxF(&m1[d * 4 + 3], lg.w);
}

// ---------------- layer-1 exp + segment sum ----------------
__global__ void k_expsum1(const int* __restrict__ ei, float* __restrict__ lg1,
                          const float* __restrict__ m1, float* __restrict__ z1, int E) {
  int e = blockIdx.x * blockDim.x + threadIdx.x;
  if (e >= E) return;
  int d = ei[E + e];
  float4 lg = *(const float4*)(lg1 + (size_t)e * 4);
  float4 m  = *(const float4*)(m1 + (size_t)d * 4);
  float4 p;
  p.x = __expf(lg.x - m.x);
  p.y = __expf(lg.y - m.y);
  p.z = __expf(lg.z - m.z);
  p.w = __expf(lg.w - m.w);
  *(float4*)(lg1 + (size_t)e * 4) = p;  // overwrite logits with p
  atomicAdd(&z1[d * 4 + 0], p.x);
  atomicAdd(&z1[d * 4 + 1], p.y);
  atomicAdd(&z1[d * 4 + 2], p.z);
  atomicAdd(&z1[d * 4 + 3], p.w);
}

// ---------------- layer-1 message scatter (one edge per 256-thread block) ----
__global__ void k_msg1(const int* __restrict__ ei, const float* __restrict__ p1,
                       const float* __restrict__ z1, const float* __restrict__ h1,
                       float* __restrict__ out1, int E) {
  int e = blockIdx.x;
  int c = threadIdx.x;  // 256 channels
  int s = ei[e], d = ei[E + e];
  int h = c >> 6;
  float alpha = p1[(size_t)e * 4 + h] / (z1[d * 4 + h] + 1e-16f);
  atomicAdd(&out1[(size_t)d * 256 + c], alpha * h1[(size_t)s * 256 + c]);
}

// ---------------- fp32 WMMA GEMM: C[M,64] = A[M,K] @ B[K,64] + bb[64] -------
// block = 128 threads = 4 waves; wave w owns the 16x16 tile (blockIdx.x, w).
// A tile staged global->LDS with GLOBAL_LOAD_ASYNC_TO_LDS_B128 (ASYNCcnt),
// LDS row stride K+4 floats: keeps 16B alignment for b128 chunks and gives
// conflict-free column reads (bank = (4*row + col) % 64, distinct per row).
// V_WMMA_F32_16X16X4_F32 layouts (ISA 7.12.2):
//   A 16x4: lane&15 = M row; VGPR v holds K = v + (lane>>4)*2
//   B 4x16: lane&15 = N col; VGPR v holds K = v + (lane>>4)*2
//   C 16x16: VGPR i -> M = i + (lane>>4)*8, N = lane&15
template <int K>
__global__ void k_gemm_wmma(const float* __restrict__ A, const float* __restrict__ B,
                            const float* __restrict__ bb, float* __restrict__ C, int M) {
  constexpr int LDA = K + 4;
  __shared__ float smA[16 * LDA];
  const int mtile = blockIdx.x;
  const int lane  = threadIdx.x & 31;
  const int ntile = threadIdx.x >> 5;  // wave id, N = 64 -> 4 tiles

  // async-stage the 16xK A tile into LDS in 16-byte chunks (uniform trip count)
  const unsigned ldsBase = (unsigned)(size_t)(&smA[0]);  // low 32 bits = LDS offset
  for (int t = threadIdx.x; t < 16 * (K / 4); t += blockDim.x) {
    int r = t / (K / 4), c4 = (t % (K / 4)) * 4;
    int rg = mtile * 16 + r;
    if (rg >= M) rg = M - 1;  // clamp tail (dup rows; store is guarded)
    const float* ga = A + (size_t)rg * K + c4;
    unsigned la = ldsBase + (unsigned)((r * LDA + c4) * 4);
    asm volatile("global_load_async_to_lds_b128 %0, %1, off"
                 :: "v"(la), "v"(ga) : "memory");
  }
  asm volatile("s_wait_asynccnt 0x0" ::: "memory");
  __syncthreads();

  const int row   = lane & 15;            // A row / B col / C col within tile
  const int khalf = (lane >> 4) * 2;
  const float* Bp = B + ntile * 16 + row;

  v8f acc = {};
#pragma unroll 4
  for (int k = 0; k < K; k += 4) {
    v2f a, b;
    a.x = smA[row * LDA + k + khalf];
    a.y = smA[row * LDA + k + khalf + 1];
    b.x = Bp[(size_t)(k + khalf) * 64];
    b.y = Bp[(size_t)(k + khalf + 1) * 64];
    acc = __builtin_amdgcn_wmma_f32_16x16x4_f32(false, a, false, b, (short)0, acc,
                                                false, false);
  }

  float* Cp = C + (size_t)(mtile * 16) * 64 + ntile * 16 + row;
  float badd = bb[ntile * 16 + row];
  if (mtile * 16 + 16 <= M) {  // uniform fast path: full tile, straight stores
#pragma unroll
    for (int i = 0; i < 8; ++i)
      Cp[(size_t)(i + (lane >> 4) * 8) * 64] = acc[i] + badd;
  } else {
#pragma unroll
    for (int i = 0; i < 8; ++i) {
      int mrow = i + (lane >> 4) * 8;
      if (mtile * 16 + mrow < M) Cp[(size_t)mrow * 64] = acc[i] + badd;
    }
  }
}

// ---------------- layer-2 node attn dots ----------------
__global__ void k_node2(const float* __restrict__ h2, const float* __restrict__ as2,
                        const float* __restrict__ ad2, float* __restrict__ als2,
                        float* __restrict__ ald2, int N) {
  int n = blockIdx.x * blockDim.x + threadIdx.x;
  if (n >= N) return;
  const float* r = h2 + (size_t)n * 64;
  float a = 0.f, b = 0.f;
#pragma unroll
  for (int i = 0; i < 64; i += 4) {
    float4 v = *(const float4*)(r + i);
    float4 s = *(const float4*)(as2 + i);
    float4 t = *(const float4*)(ad2 + i);
    a += v.x * s.x + v.y * s.y + v.z * s.z + v.w * s.w;
    b += v.x * t.x + v.y * t.y + v.z * t.z + v.w * t.w;
  }
  als2[n] = a;
  ald2[n] = b;
}

// ---------------- layer-2 edge logits / exp-sum (H=1) ----------------
__global__ void k_logits2(const int* __restrict__ ei, const float* __restrict__ ea,
                          const float* __restrict__ als2, const float* __restrict__ ald2,
                          const float* __restrict__ ce, float* __restrict__ lg2,
                          float* __restrict__ m2, int E) {
  int e = blockIdx.x * blockDim.x + threadIdx.x;
  if (e >= E) return;
  int s = ei[e], d = ei[E + e];
  float lg = leaky(als2[s] + ald2[d] + ea[e] * ce[HEADS]);
  lg2[e] = lg;
  atomicMaxF(&m2[d], lg);
}

__global__ void k_expsum2(const int* __restrict__ ei, float* __restrict__ lg2,
                          const float* __restrict__ m2, float* __restrict__ z2, int E) {
  int e = blockIdx.x * blockDim.x + threadIdx.x;
  if (e >= E) return;
  int d = ei[E + e];
  float p = __expf(lg2[e] - m2[d]);
  lg2[e] = p;
  atomicAdd(&z2[d], p);
}

// ---------------- layer-2 message scatter (4 edges x 64 ch per block) -------
__global__ void k_msg2(const int* __restrict__ ei, const float* __restrict__ p2,
                       const float* __restrict__ z2, const float* __restrict__ h2,
                       float* __restrict__ out2, int E) {
  int e = blockIdx.x * 4 + (threadIdx.x >> 6);
  if (e >= E) return;
  int c = threadIdx.x & 63;
  int s = ei[e], d = ei[E + e];
  float alpha = p2[e] / (z2[d] + 1e-16f);
  atomicAdd(&out2[(size_t)d * 64 + c], alpha * h2[(size_t)s * 64 + c]);
}

// ---------------- final edge MLP: out[e] = relu(g[s]+g[d]+br1) . Wr2 + br2 --
__global__ void k_edge_out(const int* __restrict__ ei, const float* __restrict__ g,
                           const float* __restrict__ br1, const float* __restrict__ Wr2,
                           const float* __restrict__ br2, float* __restrict__ out, int E) {
  int e = blockIdx.x * blockDim.x + threadIdx.x;
  if (e >= E) return;
  int s = ei[e], d = ei[E + e];
  const float* gs = g + (size_t)s * 64;
  const float* gd = g + (size_t)d * 64;
  float acc = 0.f;
#pragma unroll
  for (int i = 0; i < 64; i += 4) {
    float4 a  = *(const float4*)(gs + i);
    float4 b  = *(const float4*)(gd + i);
    float4 bb = *(const float4*)(br1 + i);
    float4 w  = *(const float4*)(Wr2 + i);
    float v;
    v = a.x + b.x + bb.x; acc += (v > 0.f ? v : 0.f) * w.x;
    v = a.y + b.y + bb.y; acc += (v > 0.f ? v : 0.f) * w.y;
    v = a.z + b.z + bb.z; acc += (v > 0.f ? v : 0.f) * w.z;
    v = a.w + b.w + bb.w; acc += (v > 0.f ? v : 0.f) * w.w;
  }
  out[e] = acc + br2[0];
}

extern "C" void kernel_launch(void* const* d_in, const int* in_sizes, int n_in,
                              void* d_out, int out_size, void* d_ws, size_t ws_size,
                              hipStream_t stream) {
  const float* x   = (const float*)d_in[0];
  const int*   ei  = (const int*)d_in[1];
  const float* ea  = (const float*)d_in[2];
  const float* W1  = (const float*)d_in[3];
  const float* We1 = (const float*)d_in[4];
  const float* as1 = (const float*)d_in[5];
  const float* ad1 = (const float*)d_in[6];
  const float* ae1 = (const float*)d_in[7];
  const float* b1  = (const float*)d_in[8];
  const float* W2  = (const float*)d_in[9];
  const float* We2 = (const float*)d_in[10];
  const float* as2 = (const float*)d_in[11];
  const float* ad2 = (const float*)d_in[12];
  const float* ae2 = (const float*)d_in[13];
  const float* b2  = (const float*)d_in[14];
  const float* Wr1 = (const float*)d_in[15];
  const float* br1 = (const float*)d_in[16];
  const float* Wr2 = (const float*)d_in[17];
  const float* br2 = (const float*)d_in[18];
  float* out = (float*)d_out;

  const int N = in_sizes[0] / 2;  // 100000
  const int E = in_sizes[1] / 2;  // 1600000

  // ---- workspace layout (floats), ~322 MB total ----
  float* ws = (float*)d_ws;
  size_t off = 0;
  auto alloc = [&](size_t n) {
    float* p = ws + off;
    off += (n + 3) & ~(size_t)3;
    return p;
  };
  float* h1   = alloc((size_t)N * 256);  // layer-1 node features
  float* out1 = alloc((size_t)N * 256);  // layer-1 aggregated
  float* als1 = alloc((size_t)N * 4);
  float* ald1 = alloc((size_t)N * 4);
  float* m1   = alloc((size_t)N * 4);
  float* z1   = alloc((size_t)N * 4);
  float* p1   = alloc((size_t)E * 4);    // logits -> probs
  float* h2   = alloc((size_t)N * 64);   // layer-2 transformed
  float* out2 = alloc((size_t)N * 64);   // layer-2 aggregated
  float* als2 = alloc((size_t)N);
  float* ald2 = alloc((size_t)N);
  float* m2   = alloc((size_t)N);
  float* z2   = alloc((size_t)N);
  float* p2   = alloc((size_t)E);
  float* g    = alloc((size_t)N * 64);   // h2f @ Wr1
  float* ce   = alloc(8);
  float* bb1  = alloc(64);               // b1^T @ W2
  float* bb2  = alloc(64);               // b2^T @ Wr1
  (void)ws_size;

  const int TB = 256;
  const int EB = (E + TB - 1) / TB;
  const int NB = (N + TB - 1) / TB;
  const int MT = (N + 15) / 16;

  // constants + accumulator init (every call: buffers persist across replays)
  k_ce<<<1, 32, 0, stream>>>(We1, ae1, We2, ae2, ce);
  k_biasrow<<<1, 64, 0, stream>>>(b1, W2, bb1, 256);
  k_biasrow<<<1, 64, 0, stream>>>(b2, Wr1, bb2, 64);
  k_fill<<<2048, TB, 0, stream>>>(out1, 0.f, (long long)N * 256);
  k_fill<<<512, TB, 0, stream>>>(m1, -INFINITY, (long long)N * 4);
  k_fill<<<512, TB, 0, stream>>>(z1, 0.f, (long long)N * 4);
  k_fill<<<1024, TB, 0, stream>>>(out2, 0.f, (long long)N * 64);
  k_fill<<<256, TB, 0, stream>>>(m2, -INFINITY, (long long)N);
  k_fill<<<256, TB, 0, stream>>>(z2, 0.f, (long long)N);

  // ---- layer 1 ----
  k_node1<<<N, 256, 0, stream>>>(x, W1, as1, ad1, h1, als1, ald1);
  k_logits1<<<EB, TB, 0, stream>>>(ei, ea, als1, ald1, ce, p1, m1, E);
  k_expsum1<<<EB, TB, 0, stream>>>(ei, p1, m1, z1, E);
  k_msg1<<<E, 256, 0, stream>>>(ei, p1, z1, h1, out1, E);

  // ---- layer 2: h2 = (out1 + b1) @ W2  (WMMA fp32, async-LDS staged) ----
  k_gemm_wmma<256><<<MT, 128, 0, stream>>>(out1, W2, bb1, h2, N);
  k_node2<<<NB, TB, 0, stream>>>(h2, as2, ad2, als2, ald2, N);
  k_logits2<<<EB, TB, 0, stream>>>(ei, ea, als2, ald2, ce, p2, m2, E);
  k_expsum2<<<EB, TB, 0, stream>>>(ei, p2, m2, z2, E);
  k_msg2<<<(E + 3) / 4, 256, 0, stream>>>(ei, p2, z2, h2, out2, E);

  // ---- regressor: g = (out2 + b2) @ Wr1  (WMMA fp32); then per-edge head ----
  k_gemm_wmma<64><<<MT, 128, 0, stream>>>(out2, Wr1, bb2, g, N);
  k_edge_out<<<EB, TB, 0, stream>>>(ei, g, br1, Wr2, br2, out, E);
}